// Header_13340168421796
// MI455X (gfx1250) — compile-verified
//
#include <hip/hip_runtime.h>
#include <stdint.h>

// CenterNet-style decode for MI455X (gfx1250, wave32).
// 4 phases: histogram of peak scores -> threshold -> collect -> sort+emit.
// CDNA5 path: async global->LDS staging of halo tiles + s_wait_asynccnt.

#define BB     16
#define HH     128
#define WW     128
#define NC     80
#define K_OUT  100
#define NBINS  16384
#define BIN_SHIFT 17
#define CAP    4096
#define TILE_W 64
#define TCOLS  (TILE_W + 2)      // 66 columns incl. halo
#define ROWF   (TCOLS * NC)      // 5280 floats per staged row
#define SORT_N 4096

#if defined(__HIP_DEVICE_COMPILE__) && __has_builtin(__builtin_amdgcn_global_load_async_to_lds_b128)
#define USE_ASYNC_LDS 1
typedef int v4i __attribute__((vector_size(16)));
typedef __attribute__((address_space(1))) v4i* as1_v4i_ptr;
typedef __attribute__((address_space(3))) v4i* as3_v4i_ptr;
#else
#define USE_ASYNC_LDS 0
#endif

__global__ void zero_kernel(uint32_t* __restrict__ p, int n) {
  int i = blockIdx.x * blockDim.x + threadIdx.x;
  if (i < n) p[i] = 0u;
}

// COLLECT=0: histogram peaks. COLLECT=1: append peaks >= threshold.
template <int COLLECT>
__global__ __launch_bounds__(256) void pass_kernel(
    const float* __restrict__ cls, uint32_t* __restrict__ hist,
    const uint32_t* __restrict__ thr, uint32_t* __restrict__ cnt,
    unsigned long long* __restrict__ cand) {
  __shared__ float tile[3 * ROWF];  // 63,360 bytes
  const int bx = blockIdx.x;        // (xtile, y)
  const int b  = blockIdx.y;
  const int xt = bx & 1;
  const int y  = bx >> 1;
  const int x0 = xt * TILE_W;

  // Stage rows y-1..y+1, columns x0-1..x0+TILE_W (clamped) into LDS.
  for (int r = 0; r < 3; ++r) {
    int yy = y + r - 1;
    if (yy < 0 || yy >= HH) continue;
    int xs = x0 - 1; if (xs < 0) xs = 0;
    int xe = x0 + TILE_W; if (xe > WW - 1) xe = WW - 1;  // inclusive
    int ncol = xe - xs + 1;
    int nvec = ncol * (NC / 4);  // float4 chunks (80 floats = 20 vec4 per col)
    const float* gsrc = cls + ((size_t)(b * HH + yy) * WW + xs) * NC;
    int ldst = (r * TCOLS + (xs - (x0 - 1))) * NC;  // float offset in tile
    for (int v = threadIdx.x; v < nvec; v += 256) {
#if USE_ASYNC_LDS
      __builtin_amdgcn_global_load_async_to_lds_b128(
          (as1_v4i_ptr)(gsrc + 4 * v),
          (as3_v4i_ptr)(&tile[ldst + 4 * v]),
          0, 0);
#else
      *(float4*)(&tile[ldst + 4 * v]) = *(const float4*)(gsrc + 4 * v);
#endif
    }
  }
#if USE_ASYNC_LDS
  __builtin_amdgcn_s_wait_asynccnt(0);
#endif
  __syncthreads();

  uint32_t thrB = 0u;
  if (COLLECT) thrB = thr[b];

  const int NCAND = TILE_W * NC;  // 5120 candidates per block
  for (int j = threadIdx.x; j < NCAND; j += 256) {
    int c  = j % NC;
    int xl = j / NC;       // 0..63
    int t  = xl + 1;       // tile column of center
    float center = tile[(1 * TCOLS + t) * NC + c];
    float m = center;
    for (int r = 0; r < 3; ++r) {
      int yy = y + r - 1;
      if (yy < 0 || yy >= HH) continue;
      for (int dt = -1; dt <= 1; ++dt) {
        int xg = x0 + xl + dt;
        if (xg < 0 || xg >= WW) continue;
        m = fmaxf(m, tile[(r * TCOLS + (t + dt)) * NC + c]);
      }
    }
    if (center == m) {  // peak (sigmoid strictly monotone -> logit-space NMS)
      float score = 1.0f / (1.0f + __expf(-center));
      uint32_t bits = __float_as_uint(score);  // score in (0,1): bit order == value order
      if (COLLECT) {
        if (bits >= thrB) {
          uint32_t idx = (uint32_t)(((y * WW) + (x0 + xl)) * NC + c);
          uint32_t pos = atomicAdd(&cnt[b], 1u);
          if (pos < CAP) {
            unsigned long long key =
                ((unsigned long long)bits << 32) | (unsigned long long)(~idx);
            cand[(size_t)b * CAP + pos] = key;
          }
        }
      } else {
        atomicAdd(&hist[b * NBINS + (bits >> BIN_SHIFT)], 1u);
      }
    }
  }
}

__global__ __launch_bounds__(256) void thresh_kernel(
    const uint32_t* __restrict__ hist, uint32_t* __restrict__ thr) {
  const int b = blockIdx.x;
  const uint32_t* h = hist + b * NBINS;
  __shared__ uint32_t partial[256];
  __shared__ uint32_t prefix[256];
  __shared__ int sT;
  const int t = threadIdx.x;
  const int CH = NBINS / 256;  // 64 bins per thread, chunk 0 = top bins
  int lo = NBINS - CH * (t + 1);
  uint32_t s = 0;
  for (int i = 0; i < CH; ++i) s += h[lo + i];
  partial[t] = s;
  if (t == 0) sT = 0;
  __syncthreads();
  if (t == 0) {
    uint32_t acc = 0;
    for (int i = 0; i < 256; ++i) { prefix[i] = acc; acc += partial[i]; }
  }
  __syncthreads();
  uint32_t pre = prefix[t];
  if (pre < K_OUT && pre + partial[t] >= K_OUT) {
    uint32_t cum = pre;
    for (int bin = lo + CH - 1; bin >= lo; --bin) {
      cum += h[bin];
      if (cum >= K_OUT) { sT = bin; break; }
    }
  }
  __syncthreads();
  if (t == 0) thr[b] = ((uint32_t)sT) << BIN_SHIFT;
}

__global__ __launch_bounds__(512) void final_kernel(
    const unsigned long long* __restrict__ cand, const uint32_t* __restrict__ cnt,
    const float* __restrict__ dxy, const float* __restrict__ swh,
    float* __restrict__ out) {
  __shared__ unsigned long long sk[SORT_N];  // 32 KB
  const int b = blockIdx.x;
  uint32_t n = cnt[b]; if (n > CAP) n = CAP;
  for (int i = threadIdx.x; i < SORT_N; i += 512)
    sk[i] = (i < (int)n) ? cand[(size_t)b * CAP + i] : 0ull;
  __syncthreads();
  // Bitonic sort, descending (all real keys have bits!=0 so key > 0 > pads).
  for (unsigned k = 2; k <= SORT_N; k <<= 1) {
    for (unsigned j = k >> 1; j > 0; j >>= 1) {
      for (unsigned i = threadIdx.x; i < SORT_N; i += 512) {
        unsigned ixj = i ^ j;
        if (ixj > i) {
          unsigned long long a = sk[i], c2 = sk[ixj];
          bool desc = ((i & k) == 0);
          if (desc ? (a < c2) : (a > c2)) { sk[i] = c2; sk[ixj] = a; }
        }
      }
      __syncthreads();
    }
  }
  for (int k2 = threadIdx.x; k2 < K_OUT; k2 += 512) {
    unsigned long long key = sk[k2];
    uint32_t bits = (uint32_t)(key >> 32);
    float conf = __uint_as_float(bits);
    uint32_t idx = ~((uint32_t)key);
    float x1 = 0.f, y1 = 0.f, x2 = 0.f, y2 = 0.f, cls_f = 0.f;
    if (bits != 0u) {
      uint32_t c = idx % NC;
      uint32_t sidx = idx / NC;
      uint32_t x = sidx % WW;
      uint32_t yq = sidx / WW;
      float dx = dxy[((size_t)b * HH * WW + sidx) * 2 + 0];
      float dy = dxy[((size_t)b * HH * WW + sidx) * 2 + 1];
      float sw = swh[((size_t)b * HH * WW + sidx) * 2 + 0];
      float sh = swh[((size_t)b * HH * WW + sidx) * 2 + 1];
      float xs = (float)x + dx;
      float ys = (float)yq + dy;
      float hw = sw * 0.5f, hh = sh * 0.5f;
      const float invW = 1.0f / (float)WW, invH = 1.0f / (float)HH;
      x1 = (xs - hw) * invW; y1 = (ys - hh) * invH;
      x2 = (xs + hw) * invW; y2 = (ys + hh) * invH;
      cls_f = (float)c;
    } else {
      conf = 0.0f;
    }
    size_t obase = (size_t)b * K_OUT + k2;
    out[obase * 4 + 0] = x1;
    out[obase * 4 + 1] = y1;
    out[obase * 4 + 2] = x2;
    out[obase * 4 + 3] = y2;
    out[(size_t)BB * K_OUT * 4 + obase] = conf;   // confi after boxes
    out[(size_t)BB * K_OUT * 5 + obase] = cls_f;  // classes after confi
  }
}

extern "C" void kernel_launch(void* const* d_in, const int* in_sizes, int n_in,
                              void* d_out, int out_size, void* d_ws, size_t ws_size,
                              hipStream_t stream) {
  const float* cls = (const float*)d_in[0];
  const float* dxy = (const float*)d_in[1];
  const float* swh = (const float*)d_in[2];
  float* out = (float*)d_out;

  // Workspace layout: hist[B*NBINS] | thr[B] | cnt[B] | cand[B*CAP] (u64)
  uint32_t* hist = (uint32_t*)d_ws;
  uint32_t* thr  = hist + BB * NBINS;
  uint32_t* cnt  = thr + BB;
  unsigned long long* cand = (unsigned long long*)(cnt + BB);

  int nz = BB * NBINS + 2 * BB;
  zero_kernel<<<(nz + 255) / 256, 256, 0, stream>>>(hist, nz);

  dim3 gA(2 * HH, BB);  // 2 x-tiles per row, 128 rows, 16 batches
  pass_kernel<0><<<gA, 256, 0, stream>>>(cls, hist, thr, cnt, cand);
  thresh_kernel<<<BB, 256, 0, stream>>>(hist, thr);
  pass_kernel<1><<<gA, 256, 0, stream>>>(cls, hist, thr, cnt, cand);
  final_kernel<<<BB, 512, 0, stream>>>(cand, cnt, dxy, swh, out);
}